// IDFT_19490561590114
// MI455X (gfx1250) — compile-verified
//
#include <hip/hip_runtime.h>

typedef float v2f __attribute__((ext_vector_type(2)));
typedef float v8f __attribute__((ext_vector_type(8)));
typedef int   v4i __attribute__((ext_vector_type(4)));

#define NFFT 1024
#define TM 64
#define TN 64
#define TK 32
#define SA 36   // LDS row stride (dwords): 144B rows -> 16B aligned, bank-staggered
#define SB 36
#define NCHUNK (NFFT / TK)

// ---- CDNA5 async global->LDS copy (probe via __has_builtin, safe fallback) ----
#if defined(__has_builtin)
#if __has_builtin(__builtin_amdgcn_global_load_async_to_lds_b128)
#define HAVE_ASYNC_LDS 1
#endif
#endif
#ifndef HAVE_ASYNC_LDS
#define HAVE_ASYNC_LDS 0
#endif

#if HAVE_ASYNC_LDS
typedef __attribute__((address_space(1))) v4i gv4i;
typedef __attribute__((address_space(3))) v4i sv4i;
static __device__ __forceinline__ void async_copy16(const float* g, float* l) {
  __builtin_amdgcn_global_load_async_to_lds_b128((gv4i*)g, (sv4i*)l, 0, 0);
}
static __device__ __forceinline__ void wait_async() {
#if defined(__has_builtin)
#if __has_builtin(__builtin_amdgcn_s_wait_asynccnt)
  __builtin_amdgcn_s_wait_asynccnt(0);
#else
  asm volatile("s_wait_asynccnt 0x0" ::: "memory");
#endif
#else
  asm volatile("s_wait_asynccnt 0x0" ::: "memory");
#endif
}
#endif

static __device__ __forceinline__ v8f wmma_f32(v2f a, v2f b, v8f c) {
  // D = A(16x4 f32) * B(4x16 f32) + C(16x16 f32)
  return __builtin_amdgcn_wmma_f32_16x16x4_f32(false, a, false, b, (short)0, c,
                                               false, false);
}

__global__ __launch_bounds__(256)
void idft_wmma_kernel(const float* __restrict__ Are, const float* __restrict__ Aim,
                      const float* __restrict__ Br,  const float* __restrict__ Bi,
                      float* __restrict__ Yr, float* __restrict__ Yi) {
  __shared__ float sAre[2][TM * SA];
  __shared__ float sAim[2][TM * SA];
  __shared__ float sBr[2][TN * SB];
  __shared__ float sBi[2][TN * SB];

  const int tid  = threadIdx.x;
  const int lane = tid & 31;
  const int wid  = tid >> 5;        // 0..7
  const int wm   = (wid >> 2) * 32; // wave row base inside block tile
  const int wn   = (wid & 3) * 16;  // wave col base inside block tile
  const int half = lane >> 4;
  const int l16  = lane & 15;

  const int block_m = blockIdx.y * TM;
  const int block_n = blockIdx.x * TN;

  // fill-pattern indices (j in {0,1} adds 32 A-rows / 16 B-k-rows)
  const int am = tid >> 3;          // A row 0..31
  const int ak = (tid & 7) << 2;    // A k-quad
  const int bk = tid >> 4;          // B k-row 0..15
  const int bn = (tid & 15) << 2;   // B col-quad

  v8f rr0 = {}, ii0 = {}, ri0 = {}, ir0 = {};
  v8f rr1 = {}, ii1 = {}, ri1 = {}, ir1 = {};

  float4 rBr[2], rBi[2];
#if !HAVE_ASYNC_LDS
  float4 rAr[2], rAi[2];
#endif

  // Issue global fetches for chunk at k-offset kc into LDS buffer `buf`.
  auto issue_fetch = [&](int kc, int buf) {
#pragma unroll
    for (int j = 0; j < 2; ++j) {
      const size_t ga = (size_t)(block_m + am + 32 * j) * NFFT + (size_t)(kc + ak);
#if HAVE_ASYNC_LDS
      async_copy16(Are + ga, &sAre[buf][(am + 32 * j) * SA + ak]);
      async_copy16(Aim + ga, &sAim[buf][(am + 32 * j) * SA + ak]);
#else
      rAr[j] = *(const float4*)(Are + ga);
      rAi[j] = *(const float4*)(Aim + ga);
#endif
      const size_t gb = (size_t)(kc + bk + 16 * j) * NFFT + (size_t)(block_n + bn);
      rBr[j] = *(const float4*)(Br + gb);
      rBi[j] = *(const float4*)(Bi + gb);
    }
  };

  // Publish staged data into LDS buffer `buf` (B transposed to [n][k]).
  auto publish = [&](int buf) {
#pragma unroll
    for (int j = 0; j < 2; ++j) {
#if !HAVE_ASYNC_LDS
      *(float4*)(&sAre[buf][(am + 32 * j) * SA + ak]) = rAr[j];
      *(float4*)(&sAim[buf][(am + 32 * j) * SA + ak]) = rAi[j];
#endif
      const int k = bk + 16 * j;
      sBr[buf][(bn + 0) * SB + k] = rBr[j].x;
      sBr[buf][(bn + 1) * SB + k] = rBr[j].y;
      sBr[buf][(bn + 2) * SB + k] = rBr[j].z;
      sBr[buf][(bn + 3) * SB + k] = rBr[j].w;
      sBi[buf][(bn + 0) * SB + k] = rBi[j].x;
      sBi[buf][(bn + 1) * SB + k] = rBi[j].y;
      sBi[buf][(bn + 2) * SB + k] = rBi[j].z;
      sBi[buf][(bn + 3) * SB + k] = rBi[j].w;
    }
#if HAVE_ASYNC_LDS
    wait_async();   // async A-tile copies must land before the barrier publishes
#endif
  };

  auto compute = [&](int buf) {
    const float* Ar  = sAre[buf];
    const float* Ai  = sAim[buf];
    const float* Brs = sBr[buf];
    const float* Bis = sBi[buf];
#pragma unroll
    for (int kk = 0; kk < TK; kk += 4) {
      const int kb = kk + half * 2;
      const v2f ar0 = *(const v2f*)(&Ar[(wm +      l16) * SA + kb]);
      const v2f ai0 = *(const v2f*)(&Ai[(wm +      l16) * SA + kb]);
      const v2f ar1 = *(const v2f*)(&Ar[(wm + 16 + l16) * SA + kb]);
      const v2f ai1 = *(const v2f*)(&Ai[(wm + 16 + l16) * SA + kb]);
      const v2f br  = *(const v2f*)(&Brs[(wn + l16) * SB + kb]);
      const v2f bi  = *(const v2f*)(&Bis[(wn + l16) * SB + kb]);
      rr0 = wmma_f32(ar0, br, rr0);
      ii0 = wmma_f32(ai0, bi, ii0);
      ri0 = wmma_f32(ar0, bi, ri0);
      ir0 = wmma_f32(ai0, br, ir0);
      rr1 = wmma_f32(ar1, br, rr1);
      ii1 = wmma_f32(ai1, bi, ii1);
      ri1 = wmma_f32(ar1, bi, ri1);
      ir1 = wmma_f32(ai1, br, ir1);
    }
  };

  // ---- software-pipelined main loop (double-buffered LDS) ----
  issue_fetch(0, 0);
  publish(0);
  __syncthreads();

  for (int ci = 0; ci < NCHUNK; ++ci) {
    const int cur = ci & 1;
    if (ci + 1 < NCHUNK) issue_fetch((ci + 1) * TK, cur ^ 1);  // overlaps compute
    compute(cur);
    if (ci + 1 < NCHUNK) publish(cur ^ 1);
    __syncthreads();
  }

  // ---- epilogue: y_real = rr - ii, y_imag = ri + ir ----
  const int col  = block_n + wn + l16;
  const int row0 = block_m + wm + half * 8;  // C/D layout: vgpr v -> M = v + half*8
#pragma unroll
  for (int v = 0; v < 8; ++v) {
    const size_t o0 = (size_t)(row0 + v) * NFFT + col;
    const size_t o1 = (size_t)(row0 + 16 + v) * NFFT + col;
    Yr[o0] = rr0[v] - ii0[v];
    Yi[o0] = ri0[v] + ir0[v];
    Yr[o1] = rr1[v] - ii1[v];
    Yi[o1] = ri1[v] + ir1[v];
  }
}

extern "C" void kernel_launch(void* const* d_in, const int* in_sizes, int n_in,
                              void* d_out, int out_size, void* d_ws, size_t ws_size,
                              hipStream_t stream) {
  const float* Are = (const float*)d_in[0];  // re   (8,4096,1024)
  const float* Aim = (const float*)d_in[1];  // im   (8,4096,1024)
  const float* Br  = (const float*)d_in[2];  // fft_real (1024,1024)
  const float* Bi  = (const float*)d_in[3];  // fft_imag (1024,1024)

  const int M = in_sizes[0] / NFFT;          // 32768
  float* Yr = (float*)d_out;                 // y_real flat
  float* Yi = Yr + (size_t)M * NFFT;         // y_imag flat (concatenated)

  dim3 grid(NFFT / TN, M / TM);
  idft_wmma_kernel<<<grid, 256, 0, stream>>>(Are, Aim, Br, Bi, Yr, Yi);
}